// LeakeyZOPlain_11287174053916
// MI455X (gfx1250) — compile-verified
//
#include <hip/hip_runtime.h>

// LeakyZO spiking-neuron scan for MI455X (gfx1250, wave32).
//
// Shapes: inputs [T=16, B=32, C=64, H=32, W=32] flattened; output same shape.
// Recurrence (serial over T, elementwise over B*C*H*W):
//   u = 0.5*u + x[t];  s = (u >= 1.0);  u -= s;  out[t] = s
//
// Bandwidth-bound: 268 MB of traffic -> ~11.5 us floor at 23.3 TB/s.
// Strategy: one thread owns 4 consecutive spatial floats (float4 / B128),
// preloads all 16 time-plane fragments (16 outstanding global_load_b128,
// non-temporal), then runs the 16-step scan in registers and streams 16
// non-temporal global_store_b128.

#define T_STEPS 16

typedef __attribute__((ext_vector_type(4))) float v4f;

__global__ __launch_bounds__(256) void
leaky_zo_scan_kernel(const float* __restrict__ x_in,
                     float* __restrict__ out,
                     int nvec /* spatial float4 count = B*C*H*W/4 */) {
    const int vid = blockIdx.x * blockDim.x + threadIdx.x;
    if (vid >= nvec) return;

    const v4f* __restrict__ xv = reinterpret_cast<const v4f*>(x_in);
    v4f* __restrict__ ov       = reinterpret_cast<v4f*>(out);

    // Stage 1: issue all 16 plane loads up front (8 MiB apart) so the wave
    // has 16 B128 loads in flight before any dependent compute. Non-temporal:
    // every byte is read exactly once.
    v4f xt[T_STEPS];
#pragma unroll
    for (int t = 0; t < T_STEPS; ++t) {
        xt[t] = __builtin_nontemporal_load(&xv[(size_t)t * nvec + vid]);
    }

    // Stage 2: serial LIF scan in registers, streaming spike planes out.
    v4f u = (v4f)(0.0f);
#pragma unroll
    for (int t = 0; t < T_STEPS; ++t) {
        u = 0.5f * u + xt[t];                       // leak + integrate
        v4f s;
        s.x = (u.x >= 1.0f) ? 1.0f : 0.0f;          // fire (branchless cndmask)
        s.y = (u.y >= 1.0f) ? 1.0f : 0.0f;
        s.z = (u.z >= 1.0f) ? 1.0f : 0.0f;
        s.w = (u.w >= 1.0f) ? 1.0f : 0.0f;
        u -= s;                                      // soft reset (U_TH == 1)
        __builtin_nontemporal_store(s, &ov[(size_t)t * nvec + vid]);
    }
}

extern "C" void kernel_launch(void* const* d_in, const int* in_sizes, int n_in,
                              void* d_out, int out_size, void* d_ws, size_t ws_size,
                              hipStream_t stream) {
    (void)n_in; (void)out_size; (void)d_ws; (void)ws_size;

    const float* x = (const float*)d_in[0];   // inputs  [T*B, C, H, W] fp32
    // d_in[1] (random_tangents) is unused by the reference output.
    float* out = (float*)d_out;

    const int n_total   = in_sizes[0];            // 33,554,432
    const int n_spatial = n_total / T_STEPS;      //  2,097,152
    const int nvec      = n_spatial / 4;          //    524,288 float4s

    const int block = 256;                        // 8 wave32s per block
    const int grid  = (nvec + block - 1) / block; // 2048 blocks

    leaky_zo_scan_kernel<<<grid, block, 0, stream>>>(x, out, nvec);
}